// ColorLoss_39874476376144
// MI455X (gfx1250) — compile-verified
//
#include <hip/hip_runtime.h>
#include <hip/hip_bf16.h>

typedef float v2f __attribute__((ext_vector_type(2)));
typedef float v8f __attribute__((ext_vector_type(8)));

namespace {

constexpr int W_IMG   = 768;
constexpr int TILE_R  = 128;
constexpr int TILE_C  = 64;
constexpr int HALO    = 12;                  // radius of 25-tap Gaussian
constexpr int IN_R    = TILE_R + 2 * HALO;   // 152 rows of diff needed
constexpr int IN_C    = TILE_C + 2 * HALO;   // 88 cols of diff needed
constexpr int DSTR    = 92;                  // diff row stride (16B-aligned rows)
constexpr int HSTR    = 72;                  // hblur row stride (2*72 mod 64 = 16 -> conflict-free B reads)
constexpr int NIMG    = 16 * 3;              // 48 independent planes
constexpr int NTR     = W_IMG / TILE_R;      // 6
constexpr int NTC     = W_IMG / TILE_C;      // 12
constexpr int NBLOCKS = NIMG * NTR * NTC;    // 3456
constexpr double NTOT = 16.0 * 3.0 * 768.0 * 768.0;

// Normalized Gaussian, sigma=4, radius=12 (phi[i] = exp(-(i-12)^2/32)/sum)
#define PHI_LIST {                                                         \
  0.00110988f, 0.00227733f, 0.00438967f, 0.00794865f, 0.01352113f,         \
  0.02160667f, 0.03243550f, 0.04574138f, 0.06059750f, 0.07541483f,         \
  0.08816880f, 0.09683450f, 0.09990836f, 0.09683450f, 0.08816880f,         \
  0.07541483f, 0.06059750f, 0.04574138f, 0.03243550f, 0.02160667f,         \
  0.01352113f, 0.00794865f, 0.00438967f, 0.00227733f, 0.00110988f }

// compile-time-folded weights for the horizontal (VALU) pass
constexpr float PHIC[25] = PHI_LIST;

} // namespace

// dynamically indexed copy for building the banded WMMA A-matrix fragments
__constant__ float PHI_DEV[25] = PHI_LIST;

__global__ __launch_bounds__(256)
void blur_charbonnier_kernel(const float* __restrict__ x,
                             const float* __restrict__ y,
                             float* __restrict__ partial) {
    __shared__ float Dsh[IN_R * DSTR];   // diff tile (152 x 88, stride 92) ~56 KB
    __shared__ float Hsh[IN_R * HSTR];   // h-blurred tile (152 x 64, stride 72) ~44 KB
    __shared__ float RED[256];

    const int tid = threadIdx.x;
    const int bid = blockIdx.x;
    const int tc  = bid % NTC;
    const int tr  = (bid / NTC) % NTR;
    const int img = bid / (NTC * NTR);
    const int R0  = tr * TILE_R;
    const int C0  = tc * TILE_C;

    const float* __restrict__ xb = x + (size_t)img * (W_IMG * W_IMG);
    const float* __restrict__ yb = y + (size_t)img * (W_IMG * W_IMG);

    // ---- Stage 1: diff = x - y into LDS, edge-replicate clamped, coalesced ----
    for (int idx = tid; idx < IN_R * IN_C; idx += 256) {
        int dr  = idx / IN_C;
        int dc  = idx - dr * IN_C;
        int row = min(max(R0 - HALO + dr, 0), W_IMG - 1);
        int col = min(max(C0 - HALO + dc, 0), W_IMG - 1);
        int off = row * W_IMG + col;
        Dsh[dr * DSTR + dc] = xb[off] - yb[off];
    }
    __syncthreads();

    // ---- Stage 2: horizontal 25-tap blur (VALU), 4 outputs per thread ----
    // out(hr, c) = sum_j PHIC[j] * diff(hr, c + j), c in [0,64)
    for (int g = tid; g < IN_R * (TILE_C / 4); g += 256) {
        int hr = g >> 4;
        int gc = (g & 15) * 4;
        float a0 = 0.f, a1 = 0.f, a2 = 0.f, a3 = 0.f;
        const float* drow = &Dsh[hr * DSTR + gc];
#pragma unroll
        for (int t = 0; t < 28; ++t) {
            float v = drow[t];
            if (t <= 24)           a0 = fmaf(PHIC[t],     v, a0);
            if (t >= 1 && t <= 25) a1 = fmaf(PHIC[t - 1], v, a1);
            if (t >= 2 && t <= 26) a2 = fmaf(PHIC[t - 2], v, a2);
            if (t >= 3)            a3 = fmaf(PHIC[t - 3], v, a3);
        }
        float* hrow = &Hsh[hr * HSTR + gc];
        hrow[0] = a0; hrow[1] = a1; hrow[2] = a2; hrow[3] = a3;
    }
    __syncthreads();

    // ---- Stage 3: vertical blur as banded matmul via V_WMMA_F32_16X16X4_F32 ----
    // D[16x16] = A[16x40] x B[40x16];  A[m,j] = phi[j-m] (band), B[j,n] = H[r0+j][c0+n]
    const int wave = tid >> 5;
    const int lane = tid & 31;
    const int half = lane >> 4;     // which half of the wave
    const int m    = lane & 15;

    // A fragments: 16x4 f32 layout -> lanes 0-15 hold K=4s,4s+1; lanes 16-31 hold K=4s+2,4s+3
    v2f afrag[10];
#pragma unroll
    for (int s = 0; s < 10; ++s) {
        int j0  = 4 * s + 2 * half;
        int t0  = j0 - m;
        int t0c = min(max(t0, 0), 24);
        int t1c = min(max(t0 + 1, 0), 24);
        v2f a;
        a.x = (t0 >= 0     && t0 <= 24)     ? PHI_DEV[t0c] : 0.0f;
        a.y = (t0 + 1 >= 0 && t0 + 1 <= 24) ? PHI_DEV[t1c] : 0.0f;
        afrag[s] = a;
    }

    float s_local = 0.f;
#pragma unroll
    for (int tt = 0; tt < 4; ++tt) {
        int k  = wave + 8 * tt;      // tile 0..31 over the 128x64 output region
        int r0 = (k >> 2) * 16;      // output-row base (== hblur row base for j=0)
        int c0 = (k & 3) * 16;

        v8f acc = {0.f, 0.f, 0.f, 0.f, 0.f, 0.f, 0.f, 0.f};
#pragma unroll
        for (int s = 0; s < 10; ++s) {
            int j0 = 4 * s + 2 * half;
            // B fragment: VGPR0 = rows K=4s | 4s+2, VGPR1 = rows K=4s+1 | 4s+3
            v2f b;
            b.x = Hsh[(r0 + j0)     * HSTR + c0 + m];
            b.y = Hsh[(r0 + j0 + 1) * HSTR + c0 + m];
            acc = __builtin_amdgcn_wmma_f32_16x16x4_f32(
                false, afrag[s], false, b, (short)0, acc, false, false);
        }
        // Charbonnier on the 8 outputs this lane owns (each output counted once)
#pragma unroll
        for (int v = 0; v < 8; ++v) {
            float d = acc[v];
            s_local += sqrtf(fmaf(d, d, 1.0e-6f));
        }
    }

    // ---- Stage 4: deterministic block reduction ----
    RED[tid] = s_local;
    __syncthreads();
    for (int off = 128; off > 0; off >>= 1) {
        if (tid < off) RED[tid] += RED[tid + off];
        __syncthreads();
    }
    if (tid == 0) partial[bid] = RED[0];
}

__global__ __launch_bounds__(256)
void final_reduce_kernel(const float* __restrict__ partial, float* __restrict__ out) {
    __shared__ float RED[256];
    float s = 0.f;
    for (int i = threadIdx.x; i < NBLOCKS; i += 256) s += partial[i];
    RED[threadIdx.x] = s;
    __syncthreads();
    for (int off = 128; off > 0; off >>= 1) {
        if (threadIdx.x < off) RED[threadIdx.x] += RED[threadIdx.x + off];
        __syncthreads();
    }
    if (threadIdx.x == 0) out[0] = RED[0] * (float)(1.0 / NTOT);
}

extern "C" void kernel_launch(void* const* d_in, const int* in_sizes, int n_in,
                              void* d_out, int out_size, void* d_ws, size_t ws_size,
                              hipStream_t stream) {
    (void)in_sizes; (void)n_in; (void)out_size; (void)ws_size;
    const float* x = (const float*)d_in[0];
    const float* y = (const float*)d_in[1];
    float* out     = (float*)d_out;
    float* partial = (float*)d_ws;   // NBLOCKS floats = 13.8 KB scratch

    blur_charbonnier_kernel<<<dim3(NBLOCKS), dim3(256), 0, stream>>>(x, y, partial);
    final_reduce_kernel<<<dim3(1), dim3(256), 0, stream>>>(partial, out);
}